// FuzzyLayer_78692390798053
// MI455X (gfx1250) — compile-verified
//
#include <hip/hip_runtime.h>
#include <stdint.h>

// out[b,s, i*64+j] = max(x[b,s,i], K[i,j])   (B,S,I,O) = (8,2048,64,64)
// Pure broadcast-max: bandwidth-bound (256 MiB streamed out). WMMA is
// inapplicable (no multiply-accumulate); the CDNA5 paths that matter are the
// async global->LDS copy for K, conflict-aware LDS b128 reads, and
// non-temporal b128 output stores.

namespace {
constexpr uint32_t kB = 8, kS = 2048, kI = 64, kO = 64;
constexpr uint32_t kRows   = kB * kS;                    // 16384
constexpr uint32_t kTotal4 = kRows * kI * kO / 4u;       // 16,777,216 float4s
constexpr int kBlock = 256;                              // 8 wave32s
constexpr int kGrid  = 4096;                             // grid-stride x16
}  // namespace

typedef float f32x4 __attribute__((ext_vector_type(4)));
// Exact pointee type of the gfx1250 async-LDS b128 builtins (from hipcc
// diagnostic: "__attribute__((__vector_size__(4 * sizeof(int)))) int *").
typedef int i32x4v __attribute__((__vector_size__(4 * sizeof(int))));

#define GAS __attribute__((address_space(1)))
#define LAS __attribute__((address_space(3)))

__global__ __launch_bounds__(kBlock) void fuzzy_bcast_max(
    const float* __restrict__ x, const float* __restrict__ kern,
    float* __restrict__ out) {
  // 64x64 f32 kernel matrix = 16 KiB, staged once per workgroup.
  __shared__ __align__(16) float klds[kI * kO];
  const uint32_t t = threadIdx.x;

  // ---- Stage K into LDS: 1024 float4 / 256 threads = 4 per thread ------
#if defined(__HIP_DEVICE_COMPILE__) && \
    __has_builtin(__builtin_amdgcn_global_load_async_to_lds_b128)
  // CDNA5 async copy path (ASYNCcnt-tracked): global_load_async_to_lds_b128
  #pragma unroll
  for (int r = 0; r < 4; ++r) {
    const uint32_t e = (t + (uint32_t)r * kBlock) * 4u;
    __builtin_amdgcn_global_load_async_to_lds_b128(
        (GAS i32x4v*)(kern + e), (LAS i32x4v*)(klds + e),
        /*imm offset=*/0, /*cpol=*/0);
  }
#else
  #pragma unroll
  for (int r = 0; r < 4; ++r) {
    const uint32_t e = (t + (uint32_t)r * kBlock) * 4u;
    *(f32x4*)(klds + e) = *(const f32x4*)(kern + e);
  }
#endif

#if defined(__HIP_DEVICE_COMPILE__)
  // Drain ASYNCcnt before the barrier (harmless no-op on fallback path).
  #if __has_builtin(__builtin_amdgcn_s_wait_asynccnt)
  __builtin_amdgcn_s_wait_asynccnt(0);
  #else
  asm volatile("s_wait_asynccnt 0" ::: "memory");
  #endif
#endif
  __syncthreads();

  // ---- Streaming compute: one float4 of output per lane per iteration --
  // g -> (row, i, j4): lanes 0..15 share one x scalar; stores are perfectly
  // coalesced b128; LDS read sweeps banks j4*4..j4*4+3 (full 64-bank sweep
  // per half-wave).
  const uint32_t stride = (uint32_t)gridDim.x * (uint32_t)kBlock;
  uint32_t g = (uint32_t)blockIdx.x * (uint32_t)kBlock + t;
  #pragma unroll 4
  for (; g < kTotal4; g += stride) {
    const uint32_t j4  = g & 15u;
    const uint32_t i   = (g >> 4) & 63u;
    const uint32_t row = g >> 10;

    const float xv = x[row * kI + i];                       // L2-resident
    const f32x4 kv = *(const f32x4*)(klds + i * kO + j4 * 4u);

    f32x4 o;
    o.x = fmaxf(xv, kv.x);
    o.y = fmaxf(xv, kv.y);
    o.z = fmaxf(xv, kv.z);
    o.w = fmaxf(xv, kv.w);

    // Write-once 256 MiB stream: non-temporal b128 store, don't pollute
    // WGP$/L2 with dead output lines.
    __builtin_nontemporal_store(o, (f32x4*)out + g);
  }
}

extern "C" void kernel_launch(void* const* d_in, const int* /*in_sizes*/,
                              int /*n_in*/, void* d_out, int /*out_size*/,
                              void* /*d_ws*/, size_t /*ws_size*/,
                              hipStream_t stream) {
  const float* x    = (const float*)d_in[0];   // (8,2048,64) f32
  const float* kern = (const float*)d_in[1];   // (64,64) f32
  float* out        = (float*)d_out;           // (8,2048,4096) f32
  fuzzy_bcast_max<<<dim3(kGrid), dim3(kBlock), 0, stream>>>(x, kern, out);
}